// LGFFLoss_91319594648194
// MI455X (gfx1250) — compile-verified
//
#include <hip/hip_runtime.h>
#include <hip/hip_bf16.h>
#include <math.h>

// CDNA5 / gfx1250, wave32.
typedef __attribute__((ext_vector_type(2))) float v2f;
typedef __attribute__((ext_vector_type(8))) float v8f;

#define W_RATE 0.015f
#define WAVES_PER_BLOCK 8
#define CHUNK 2048            // targets staged per LDS chunk (32 KB)

__global__ void lgff_zero_out(float* out) { out[0] = 0.0f; }

// One wave32 owns a 16-row strip of pred points.
// Pairwise-d2 16x16 tile via V_WMMA_F32_16X16X4_F32:
//   A row m = (-2px, -2py, -2pz, 1)          (pred point, K padded 3->4)
//   B col n = (qx, qy, qz, |q|^2)            (target point)
//   D       = |q|^2 - 2 p.q                  (add row-constant |p|^2 after min)
// D layout: VGPR v, lanes 0-15 -> M=v, N=lane; lanes 16-31 -> M=v+8, N=lane-16.
// A layout: lanes 0-15 hold (K=0,K=1) in 2 VGPRs, lanes 16-31 hold (K=2,K=3).
// B layout (mirrored): lanes = N, VGPR+half = K.
// LDS staging is pre-split by lane-half so the inner loop is a single
// conflict-free ds_load_b64 per tile (no per-tile component selects):
//   sB[tile*32 + sub]      = (qx, qy)    -> lanes 0-15  (K=0,1)
//   sB[tile*32 + 16 + sub] = (qz, bb)    -> lanes 16-31 (K=2,3)
__global__ __launch_bounds__(256) void lgff_loss_kernel(
    const float* __restrict__ pred_quat,   // (B,N,4)
    const float* __restrict__ pred_trans,  // (B,N,3)
    const float* __restrict__ pred_conf,   // (B,N,1)
    const float* __restrict__ pose,        // (B,3,4)
    const float* __restrict__ points,      // (B,N,3)
    const int*   __restrict__ cls_id,      // (B,) int64 in memory; low word of [0] suffices
    float* __restrict__ out,
    int B, int N, float inv_total)
{
    __shared__ float2 sB[(CHUNK / 16) * 32];   // 32 KB

    const int tid  = threadIdx.x;
    const int lane = tid & 31;
    const int wave = tid >> 5;
    const int sub  = lane & 15;   // row within strip (both halves duplicate)
    const int half = lane >> 4;   // K-half selector for A/B operands

    const int blocksPerBatch = gridDim.x / B;
    const int b        = blockIdx.x / blocksPerBatch;
    const int stripIdx = (blockIdx.x % blocksPerBatch) * WAVES_PER_BLOCK + wave;
    const int rowBase  = stripIdx * 16;
    const int row      = rowBase + sub;

    // ---- per-row pred transform (redundantly in both lane-halves) ----
    const float* P = pose + b * 12;
    const float r00=P[0], r01=P[1], r02=P[2],  t0=P[3];
    const float r10=P[4], r11=P[5], r12=P[6],  t1=P[7];
    const float r20=P[8], r21=P[9], r22=P[10], t2=P[11];

    const long long base = (long long)b * N + row;
    float qw = pred_quat[base*4+0], qx = pred_quat[base*4+1];
    float qy = pred_quat[base*4+2], qz = pred_quat[base*4+3];
    float qn = sqrtf(qw*qw + qx*qx + qy*qy + qz*qz);
    qn = fmaxf(qn, 1e-8f);
    const float inq = 1.0f / qn;
    qw *= inq; qx *= inq; qy *= inq; qz *= inq;
    const float m00=1.f-2.f*(qy*qy+qz*qz), m01=2.f*(qx*qy-qw*qz), m02=2.f*(qx*qz+qw*qy);
    const float m10=2.f*(qx*qy+qw*qz), m11=1.f-2.f*(qx*qx+qz*qz), m12=2.f*(qy*qz-qw*qx);
    const float m20=2.f*(qx*qz-qw*qy), m21=2.f*(qy*qz+qw*qx), m22=1.f-2.f*(qx*qx+qy*qy);

    const float tx = points[base*3+0], ty = points[base*3+1], tz = points[base*3+2];
    const float c0 = tx - t0, c1 = ty - t1, c2 = tz - t2;
    const float pm0 = r00*c0 + r01*c1 + r02*c2;
    const float pm1 = r10*c0 + r11*c1 + r12*c2;
    const float pm2 = r20*c0 + r21*c1 + r22*c2;
    const float pp0 = m00*pm0 + m01*pm1 + m02*pm2 + pred_trans[base*3+0];
    const float pp1 = m10*pm0 + m11*pm1 + m12*pm2 + pred_trans[base*3+1];
    const float pp2 = m20*pm0 + m21*pm1 + m22*pm2 + pred_trans[base*3+2];
    const float aa  = pp0*pp0 + pp1*pp1 + pp2*pp2;

    const bool symmetric = (cls_id[0] == 1);   // uniform across the block

    float rowLossVal = 0.0f;

    if (symmetric) {
        // A operand: half 0 -> (-2px, -2py), half 1 -> (-2pz, 1)
        v2f A;
        A.x = half ? (-2.0f * pp2) : (-2.0f * pp0);
        A.y = half ? 1.0f          : (-2.0f * pp1);

        v8f rmin;
        #pragma unroll
        for (int i = 0; i < 8; ++i) rmin[i] = 3.4e38f;

        for (int cbase = 0; cbase < N; cbase += CHUNK) {
            const int cn = (N - cbase) < CHUNK ? (N - cbase) : CHUNK;
            __syncthreads();   // protect prior chunk reads
            for (int i = tid; i < cn; i += blockDim.x) {
                const long long gi = ((long long)b * N + cbase + i) * 3;
                const float x = points[gi+0], y = points[gi+1], z = points[gi+2];
                const int tile = i >> 4, s = i & 15;
                sB[tile * 32 + s]      = make_float2(x, y);
                sB[tile * 32 + 16 + s] = make_float2(z, x*x + y*y + z*z);
                // warm L2/near caches for the next chunk (global_prefetch_b8)
                if (cbase + CHUNK + i < N)
                    __builtin_prefetch(points + ((long long)b * N + cbase + CHUNK + i) * 3, 0, 0);
            }
            __syncthreads();

            const int tiles = cn >> 4;
            for (int t = 0; t < tiles; ++t) {
                const float2 q = sB[t * 32 + lane];   // one conflict-free b64 / tile
                v2f Bv;
                Bv.x = q.x;
                Bv.y = q.y;
                v8f cz = {};
                v8f d = __builtin_amdgcn_wmma_f32_16x16x4_f32(
                    /*neg_a=*/false, A, /*neg_b=*/false, Bv,
                    /*c_mod=*/(short)0, cz,
                    /*reuse_a=*/false, /*reuse_b=*/false);
                #pragma unroll
                for (int i = 0; i < 8; ++i) rmin[i] = fminf(rmin[i], d[i]);
            }
        }

        // Row-min across the 16 lanes of each half (stays inside half groups).
        #pragma unroll
        for (int i = 0; i < 8; ++i) {
            float v = rmin[i];
            v = fminf(v, __shfl_xor(v, 1, 32));
            v = fminf(v, __shfl_xor(v, 2, 32));
            v = fminf(v, __shfl_xor(v, 4, 32));
            v = fminf(v, __shfl_xor(v, 8, 32));
            rmin[i] = v;
        }

        // lanes 0..7 finalize rows rowBase+k, lanes 16..23 rows rowBase+8+k.
        const int k = lane & 7;
        float e = rmin[0];
        if (k == 1) e = rmin[1];
        if (k == 2) e = rmin[2];
        if (k == 3) e = rmin[3];
        if (k == 4) e = rmin[4];
        if (k == 5) e = rmin[5];
        if (k == 6) e = rmin[6];
        if (k == 7) e = rmin[7];

        int myRow = -1;
        if (lane < 8)                       myRow = rowBase + k;
        else if (lane >= 16 && lane < 24)   myRow = rowBase + 8 + k;

        const int srcLane = (myRow >= 0) ? (myRow - rowBase) : 0;  // lanes 0-15 hold aa[rowBase+sub]
        const float aaRow = __shfl(aa, srcLane, 32);

        if (myRow >= 0) {
            float d2 = e + aaRow;
            d2 = fmaxf(d2, 1e-12f);
            const float dist = sqrtf(d2);
            float cf = pred_conf[(long long)b * N + myRow];
            cf = fminf(fmaxf(cf, 1e-4f), 1.0f);
            rowLossVal = (dist * cf - W_RATE * logf(cf)) * inv_total;
        }
    } else {
        // Non-symmetric: per-index distance |pred - target| (half 0 only).
        if (half == 0) {
            const float dx = pp0 - tx, dy = pp1 - ty, dz = pp2 - tz;
            const float dist = sqrtf(dx*dx + dy*dy + dz*dz);
            float cf = pred_conf[base];
            cf = fminf(fmaxf(cf, 1e-4f), 1.0f);
            rowLossVal = (dist * cf - W_RATE * logf(cf)) * inv_total;
        }
    }

    // Wave sum (32 lanes), then one atomic per wave.
    float s = rowLossVal;
    s += __shfl_xor(s, 1, 32);
    s += __shfl_xor(s, 2, 32);
    s += __shfl_xor(s, 4, 32);
    s += __shfl_xor(s, 8, 32);
    s += __shfl_xor(s, 16, 32);
    if (lane == 0) atomicAdd(out, s);
}

extern "C" void kernel_launch(void* const* d_in, const int* in_sizes, int n_in,
                              void* d_out, int out_size, void* d_ws, size_t ws_size,
                              hipStream_t stream) {
    const float* pred_quat  = (const float*)d_in[0];
    const float* pred_trans = (const float*)d_in[1];
    const float* pred_conf  = (const float*)d_in[2];
    const float* pose       = (const float*)d_in[3];
    const float* points     = (const float*)d_in[4];
    const int*   cls_id     = (const int*)d_in[5];

    const int B = in_sizes[3] / 12;                          // pose is (B,3,4)
    const int N = (int)((long long)in_sizes[4] / (3LL * B)); // points is (B,N,3)
    const float inv_total = 1.0f / (float)((long long)B * (long long)N);

    float* out = (float*)d_out;
    lgff_zero_out<<<1, 1, 0, stream>>>(out);

    const int blocksPerBatch = N / (16 * WAVES_PER_BLOCK);   // N=4096 -> 32
    dim3 grid(B * blocksPerBatch);
    lgff_loss_kernel<<<grid, 256, 0, stream>>>(
        pred_quat, pred_trans, pred_conf, pose, points, cls_id,
        out, B, N, inv_total);
}